// SigmaBlock_70205535421261
// MI455X (gfx1250) — compile-verified
//
#include <hip/hip_runtime.h>

typedef __attribute__((ext_vector_type(2))) float v2f;
typedef __attribute__((ext_vector_type(8))) float v8f;

#define PP      4950        // P = N*(N-1)/2
#define BB      4
#define FF      8
#define HH      256
#define MM      161700      // C(100,3)
#define ROWS    (BB * MM)   // 646800, divisible by 16
#define TILES   (ROWS / 16) // 40425 waves of work

// ---------------------------------------------------------------------------
// Kernel 1: zero the 392 MB output with 128-bit stores (the bandwidth floor:
// ~392MB / 23.3 TB/s ~= 17 us dominates everything else in this problem).
// ---------------------------------------------------------------------------
__global__ __launch_bounds__(256)
void sigma_zero_kernel(float4* __restrict__ out, long n4) {
    long i = (long)blockIdx.x * blockDim.x + threadIdx.x;
    long stride = (long)gridDim.x * blockDim.x;
    float4 z; z.x = 0.f; z.y = 0.f; z.z = 0.f; z.w = 0.f;
    for (; i < n4; i += stride) out[i] = z;
}

// Single-instruction ReLU: v_med3_f32(x, 0, +inf) avoids the 2-op
// canonicalize+max sequence fmaxf() generates under IEEE rules.
__device__ __forceinline__ float relu1(float x) {
    return __builtin_amdgcn_fmed3f(x, 0.0f, __builtin_inff());
}

// ---------------------------------------------------------------------------
// Kernel 2: fused MLP (WMMA f32 16x16x4) + tanh/cospi + symmetric scatter.
// One wave32 = 16 rows of err. For each of 16 hidden column-tiles:
//   C(v)[lane] = b_h[n]           (C layout: VGPR v, lanes0-15: M=v,N=ln;
//                                             lanes16-31: M=v+8)
//   C = A(K0..3) x B + C ; C = A(K4..7) x B + C   (two v_wmma_f32_16x16x4_f32)
//   h = relu(C); s{1,2,3}[v] += h*W{1,2,3}[n]     (per-lane partial dots)
// Then shfl_xor reduce over the 16-lane half, tanh/cospi, plain-store scatter.
// Scatter is provably collision-free: an ordered pair of pair-indices sharing
// one point uniquely determines the triplet, and each triplet emits only the
// directed edges (ij->jk),(jk->ki),(ki->ij); the transposed store covers
// Sigma^T. So plain stores over a zeroed output are exact.
// ---------------------------------------------------------------------------
__global__ __launch_bounds__(256)
void sigma_mlp_scatter_kernel(const float* __restrict__ err,
                              const float* __restrict__ Wh,
                              const float* __restrict__ bh,
                              const float* __restrict__ W1, const float* __restrict__ b1,
                              const float* __restrict__ W2, const float* __restrict__ b2,
                              const float* __restrict__ W3, const float* __restrict__ b3,
                              const int*   __restrict__ ij,
                              const int*   __restrict__ jk,
                              const int*   __restrict__ ki,
                              float* __restrict__ out) {
    const int lane = threadIdx.x & 31;
    const int tile = blockIdx.x * (blockDim.x >> 5) + (threadIdx.x >> 5);
    if (tile >= TILES) return;              // wave-uniform; EXEC stays all-1s

    const int half = lane >> 4;             // 0: lanes 0-15, 1: lanes 16-31
    const int ln   = lane & 15;
    const int r0   = tile * 16;

    // A fragments: ISA 32-bit A 16x4 layout.
    //   VGPR0: K = 2*half + 0 ; VGPR1: K = 2*half + 1 ; M = ln
    const float* rowp = err + (long)(r0 + ln) * FF;
    const float2 eA = *(const float2*)(rowp + 2 * half);      // K-block 0..3
    const float2 eB = *(const float2*)(rowp + 4 + 2 * half);  // K-block 4..7
    v2f a0; a0[0] = eA.x; a0[1] = eA.y;
    v2f a1; a1[0] = eB.x; a1[1] = eB.y;

    float s1[8], s2[8], s3[8];
#pragma unroll
    for (int v = 0; v < 8; ++v) { s1[v] = 0.f; s2[v] = 0.f; s3[v] = 0.f; }

    for (int nt = 0; nt < 16; ++nt) {
        const int n = nt * 16 + ln;         // this lane's hidden column
        // B fragments: lane = N, VGPR p: K = 2*half + p (4x16 B tile)
        v2f fb0, fb1;
        fb0[0] = Wh[(2 * half + 0) * HH + n];
        fb0[1] = Wh[(2 * half + 1) * HH + n];
        fb1[0] = Wh[(4 + 2 * half + 0) * HH + n];
        fb1[1] = Wh[(4 + 2 * half + 1) * HH + n];

        const float bias = bh[n];
        v8f c;
#pragma unroll
        for (int v = 0; v < 8; ++v) c[v] = bias;

        // D = A*B + C, full f32 path (no precision loss vs reference)
        c = __builtin_amdgcn_wmma_f32_16x16x4_f32(false, a0, false, fb0,
                                                  (short)0, c, false, false);
        c = __builtin_amdgcn_wmma_f32_16x16x4_f32(false, a1, false, fb1,
                                                  (short)0, c, false, false);

        const float w1 = W1[n], w2 = W2[n], w3 = W3[n];
#pragma unroll
        for (int v = 0; v < 8; ++v) {
            const float h = relu1(c[v]);           // single v_med3_f32
            s1[v] = fmaf(h, w1, s1[v]);
            s2[v] = fmaf(h, w2, s2[v]);
            s3[v] = fmaf(h, w3, s3[v]);
        }
    }

    // Reduce over the 16 lanes of each half (masks 8..1 never cross bit4).
#pragma unroll
    for (int m = 8; m >= 1; m >>= 1) {
#pragma unroll
        for (int v = 0; v < 8; ++v) {
            s1[v] += __shfl_xor(s1[v], m, 32);
            s2[v] += __shfl_xor(s2[v], m, 32);
            s3[v] += __shfl_xor(s3[v], m, 32);
        }
    }

    const float bb1 = b1[0], bb2 = b2[0], bb3 = b3[0];
    if (ln == 0) {                          // lanes 0 and 16: 8 rows each
#pragma unroll
        for (int v = 0; v < 8; ++v) {
            const int row = r0 + half * 8 + v;
            const int b   = row / MM;
            const int m   = row - b * MM;
            const float p1 = cospif(tanhf(s1[v] + bb1));
            const float p2 = cospif(tanhf(s2[v] + bb2));
            const float p3 = cospif(tanhf(s3[v] + bb3));
            const int i1 = ij[m], i2 = jk[m], i3 = ki[m];
            float* ob = out + (long)b * ((long)PP * PP);
            ob[(long)i1 * PP + i2] = p1;    // Sigma[r,c] ...
            ob[(long)i2 * PP + i1] = p1;    // ... + Sigma^T contribution
            ob[(long)i2 * PP + i3] = p2;
            ob[(long)i3 * PP + i2] = p2;
            ob[(long)i3 * PP + i1] = p3;
            ob[(long)i1 * PP + i3] = p3;
        }
    }
}

// ---------------------------------------------------------------------------
extern "C" void kernel_launch(void* const* d_in, const int* in_sizes, int n_in,
                              void* d_out, int out_size, void* d_ws, size_t ws_size,
                              hipStream_t stream) {
    const float* err = (const float*)d_in[0];
    const float* Wh  = (const float*)d_in[1];
    const float* bh  = (const float*)d_in[2];
    const float* W1  = (const float*)d_in[3];
    const float* b1  = (const float*)d_in[4];
    const float* W2  = (const float*)d_in[5];
    const float* b2  = (const float*)d_in[6];
    const float* W3  = (const float*)d_in[7];
    const float* b3  = (const float*)d_in[8];
    const int*   ij  = (const int*)d_in[9];
    const int*   jk  = (const int*)d_in[10];
    const int*   ki  = (const int*)d_in[11];
    float* out = (float*)d_out;

    // out_size = 4 * 4950 * 4950 = 98,010,000 (divisible by 4)
    const long n4 = (long)out_size / 4;
    const int zblocks = (int)((n4 + 255) / 256);
    sigma_zero_kernel<<<zblocks, 256, 0, stream>>>((float4*)out, n4);

    const int wavesPerBlock = 8;            // 256 threads, wave32
    const int blocks = (TILES + wavesPerBlock - 1) / wavesPerBlock;
    sigma_mlp_scatter_kernel<<<blocks, 256, 0, stream>>>(
        err, Wh, bh, W1, b1, W2, b2, W3, b3, ij, jk, ki, out);
}